// GreedyStructureLearner_78769700208720
// MI455X (gfx1250) — compile-verified
//
#include <hip/hip_runtime.h>
#include <hip/hip_bf16.h>
#include <math.h>

#define TOPK 16
#define NEG_INF_F (-1000000000.0f)
#define CONSUMED_F (-3.0e38f)

typedef __attribute__((ext_vector_type(2))) float        v2f;
typedef __attribute__((ext_vector_type(8))) float        v8f;
typedef __attribute__((ext_vector_type(4))) unsigned int v4u;
typedef __attribute__((ext_vector_type(8))) int          v8i;
typedef __attribute__((ext_vector_type(4))) int          v4i;

#if __has_builtin(__builtin_amdgcn_tensor_load_to_lds) && \
    __has_builtin(__builtin_amdgcn_s_wait_tensorcnt)
#define USE_TDM 1
#else
#define USE_TDM 0
#endif

// LDS row pitch for the feature tile: 256 + 2 pad DWORDs (TDM pad feature)
#define TILE_PITCH 258

// ---------------------------------------------------------------------------
// Kernel 1: importance[b, n] = dot(features[b, n, :], w[:, 0]) via
// V_WMMA_F32_16X16X4_F32 (exact fp32). One wave32 block computes a 16-row
// tile for one batch. The 16x256-f32 feature tile is DMA'd into LDS by the
// Tensor Data Mover (2-D descriptor, row pitch F*4 bytes, +2 DWORD pad per
// row -> 258-float LDS pitch, breaking the same-bank pattern), then the WMMA
// loop consumes it. B-matrix holds w[k] replicated across all 16 columns so
// every accumulator column equals the importance vector.
//
// fp32 A 16x4 layout (ISA 7.12.2): lane l holds row M=l%16;
//   VGPR0 = K=0 (lanes 0-15) / K=2 (lanes 16-31); VGPR1 = K=1 / K=3.
// fp32 B 4x16 mirrored (column j=l%16 per lane, same K split).
// ---------------------------------------------------------------------------
__global__ __launch_bounds__(32)
void importance_wmma_kernel(const float* __restrict__ feat,   // [B, N, F]
                            const float* __restrict__ w,      // [F]
                            float* __restrict__ imp,          // [B, N]
                            int N, int F) {
  const int bb      = blockIdx.x >> 8;          // batch (N/16 == 256 tiles)
  const int rowBase = (blockIdx.x & 255) << 4;  // 16-row tile
  const int lane    = threadIdx.x;
  const int m       = lane & 15;
  const int koff    = (lane < 16) ? 0 : 2;

#if USE_TDM
  __shared__ float tile[16 * TILE_PITCH];
  {
    const unsigned long long ga =
        (unsigned long long)(const void*)(feat + ((size_t)bb * N + rowBase) * F);
    const unsigned int la = (unsigned int)(size_t)(void*)tile;  // LDS byte offset

    // D# group 0: count=1 | lds_addr | global_addr[56:0] | type=2
    v4u g0 = { 1u,
               la,
               (unsigned int)ga,
               (unsigned int)((ga >> 32) & 0x1FFFFFFu) | (2u << 30) };
    // D# group 1:
    //  w0: data_size=2(4B) | pad_enable | pad_interval=7(256 DW) | pad_amount=1(2 DW)
    //  w1[31:16]=tensor_dim0.lo16   w2[31:16]=tensor_dim1.lo16 (=16 rows)
    //  w3[31:16]=tile_dim0 (=F)     w4[15:0]=tile_dim1 (=16)
    //  w5=tensor_dim0_stride.lo32 (=F elements)
    v8i g1 = { (int)((2u << 16) | (1u << 20) | (7u << 22) | (1u << 25)),
               (int)(((unsigned int)F & 0xFFFFu) << 16),
               (int)(16u << 16) | (int)(((unsigned int)F >> 16) & 0xFFFFu),
               (int)(((unsigned int)F & 0xFFFFu) << 16),
               16,
               F,
               0, 0 };
    v4i g2 = {0, 0, 0, 0};   // 2-D tensor: groups 2/3 unused
    v4i g3 = {0, 0, 0, 0};
    v8i g4 = {0, 0, 0, 0, 0, 0, 0, 0};  // clang-23 6-arg form: extra group, zero
    __builtin_amdgcn_tensor_load_to_lds(g0, g1, g2, g3, g4, 0);
    __builtin_amdgcn_s_wait_tensorcnt(0);
  }
  const float* frow = &tile[m * TILE_PITCH];
#else
  const float* frow = feat + ((size_t)bb * N + rowBase + m) * F;
#endif

  v8f acc = {};
  for (int k0 = 0; k0 < F; k0 += 4) {
    v2f a, b;
    a.x = frow[k0 + koff];
    a.y = frow[k0 + koff + 1];
    b.x = w[k0 + koff];
    b.y = w[k0 + koff + 1];
    // 8-arg form: (neg_a, A, neg_b, B, c_mod, C, reuse_a, reuse_b)
    acc = __builtin_amdgcn_wmma_f32_16x16x4_f32(
        false, a, false, b, (short)0, acc, false, false);
  }

  // C/D layout: lane l, VGPR r -> (M = r + 8*(l>=16), Ncol = l%16). Columns
  // are identical; lanes with Ncol==0 write their 8 rows.
  if (m == 0) {
    const int half = lane >> 4;
    float* dst = imp + (size_t)bb * N + rowBase + 8 * half;
#pragma unroll
    for (int r = 0; r < 8; ++r) dst[r] = acc[r];
  }
}

// ---------------------------------------------------------------------------
// Kernel 2: per row i (one block of 256 threads = 8 wave32), loop all B
// batches so the adj row is read from HBM exactly once (kept in registers).
// Thread t owns j = t + 256*c, c = 0..15 (fully coalesced loads).
// 16 rounds of (value, index) argmax; ties -> smaller index (matches
// jax.lax.top_k). Winners come out in descending order -> softmax directly.
// ---------------------------------------------------------------------------
__global__ __launch_bounds__(256)
void topk_softmax_kernel(const int* __restrict__ adj,   // [N, N]
                         const float* __restrict__ imp, // [B, N]
                         float* __restrict__ outW,      // [B, N, 16]
                         float* __restrict__ outI,      // [B, N, 16] (float-cast idx)
                         int B, int N) {
  const int row  = blockIdx.x;
  const int t    = threadIdx.x;
  const int lane = t & 31;
  const int wid  = t >> 5;

  // adj row: 16 coalesced loads, reused for all 8 batches from registers.
  int a[16];
  const int* arow = adj + (size_t)row * N;
#pragma unroll
  for (int c = 0; c < 16; ++c) a[c] = arow[t + 256 * c];

  __shared__ float sVal[8];
  __shared__ int   sIdx[8];
  __shared__ float topV[TOPK];
  __shared__ int   topI[TOPK];
  __shared__ float winV;
  __shared__ int   winI;

  for (int b = 0; b < B; ++b) {
    const float* impb = imp + (size_t)b * N;
    const float impi = impb[row];

    float s[16];
#pragma unroll
    for (int c = 0; c < 16; ++c) {
      const int j = t + 256 * c;
      s[c] = (a[c] > 0) ? (impi + impb[j]) : NEG_INF_F;
    }

    // local argmax over my 16 slots (strict > keeps smallest j on ties)
    float bv = s[0];
    int bc = 0;
#pragma unroll
    for (int c = 1; c < 16; ++c)
      if (s[c] > bv) { bv = s[c]; bc = c; }
    int bi = t + 256 * bc;

    for (int r = 0; r < TOPK; ++r) {
      // wave32 (value, index) argmax
      float v = bv;
      int   i = bi;
#pragma unroll
      for (int off = 16; off >= 1; off >>= 1) {
        const float ov = __shfl_xor(v, off, 32);
        const int   oi = __shfl_xor(i, off, 32);
        if (ov > v || (ov == v && oi < i)) { v = ov; i = oi; }
      }
      if (lane == 0) { sVal[wid] = v; sIdx[wid] = i; }
      __syncthreads();
      if (t == 0) {
        float gv = sVal[0];
        int   gi = sIdx[0];
#pragma unroll
        for (int ww = 1; ww < 8; ++ww)
          if (sVal[ww] > gv || (sVal[ww] == gv && sIdx[ww] < gi)) {
            gv = sVal[ww]; gi = sIdx[ww];
          }
        topV[r] = gv; topI[r] = gi;
        winV = gv;   winI = gi;
      }
      __syncthreads();
      // unique winner (indices are unique per thread) invalidates + rescans
      if (bv == winV && bi == winI) {
        s[bc] = CONSUMED_F;
        bv = s[0]; bc = 0;
#pragma unroll
        for (int c = 1; c < 16; ++c)
          if (s[c] > bv) { bv = s[c]; bc = c; }
        bi = t + 256 * bc;
      }
    }
    __syncthreads();

    // softmax over the 16 winners (topV is descending -> topV[0] is the max)
    if (t < TOPK) {
      const float e = __expf(topV[t] - topV[0]);
      float sum = e;
#pragma unroll
      for (int off = 8; off >= 1; off >>= 1) sum += __shfl_xor(sum, off, 32);
      const size_t o = ((size_t)b * N + row) * TOPK + t;
      outW[o] = e / sum;
      outI[o] = (float)topI[t];
    }
    __syncthreads();
  }
}

extern "C" void kernel_launch(void* const* d_in, const int* in_sizes, int n_in,
                              void* d_out, int out_size, void* d_ws, size_t ws_size,
                              hipStream_t stream) {
  // setup_inputs() order: adj [N,N] int32, features [B,N,F] f32, attn_kernel [F,1] f32
  const int*   adj  = (const int*)d_in[0];
  const float* feat = (const float*)d_in[1];
  const float* w    = (const float*)d_in[2];

  const int F = in_sizes[2];                       // 256
  int N = 1;
  while ((long long)(N + 1) * (N + 1) <= (long long)in_sizes[0]) ++N;  // N = 4096
  const int B = in_sizes[1] / (N * F);             // 8

  float* imp  = (float*)d_ws;                      // B*N floats (128 KB)
  float* outW = (float*)d_out;                     // [B,N,16] weights
  float* outI = outW + (size_t)B * N * TOPK;       // [B,N,16] indices as float

  importance_wmma_kernel<<<B * (N / 16), 32, 0, stream>>>(feat, w, imp, N, F);
  topk_softmax_kernel<<<N, 256, 0, stream>>>(adj, imp, outW, outI, B, N);
}